// FreeCSL_59880434041334
// MI455X (gfx1250) — compile-verified
//
#include <hip/hip_runtime.h>
#include <math.h>

typedef float v2f __attribute__((ext_vector_type(2)));
typedef float v8f __attribute__((ext_vector_type(8)));

#define N_PTS   8192
#define F_INN   512
#define F_HIDD  512
#define F_OUTT  256
#define N_CLUU  10
#define K_NEI   10
#define MASKW   (N_PTS / 32)   // 256 mask words per row

// ---- workspace layout (units of 4-byte elements) ----
#define OFF_SQ     ((size_t)0)          // 8192 f32
#define OFF_DEG    ((size_t)8192)       // 8192 f32
#define OFF_INV    ((size_t)16384)      // 8192 f32
#define OFF_KNN    ((size_t)24576)      // 8192*10 i32
#define OFF_MASK   ((size_t)106496)     // 8192*256 u32 (8 MB)
#define OFF_ACC    ((size_t)2203648)    // 16 f32: [0]=traceGP [1]=nEdges [2..11]=m[c]
#define OFF_T1     ((size_t)2203664)    // 8192*512 f32
#define OFF_H      ((size_t)6397968)    // 8192*512 f32
#define OFF_T2     ((size_t)10592272)   // 8192*256 f32
#define OFF_EMB    ((size_t)12689424)   // 8192*256 f32
#define OFF_ASSIGN ((size_t)14786576)   // 8192*10 f32

// -------------------- zero scratch --------------------
__global__ void zero_kernel(unsigned* __restrict__ p, size_t n) {
  size_t i = (size_t)blockIdx.x * blockDim.x + threadIdx.x;
  if (i < n) p[i] = 0u;
}

// -------------------- row squared norms --------------------
__global__ __launch_bounds__(256) void sq_kernel(const float* __restrict__ X,
                                                 float* __restrict__ sq) {
  int i = blockIdx.x * 256 + threadIdx.x;
  if (i >= N_PTS) return;
  const float* x = X + (size_t)i * F_INN;
  float s = 0.f;
  for (int k = 0; k < F_INN; ++k) s = fmaf(x[k], x[k], s);
  sq[i] = s;
}

// -------------------- fused gram + top-k --------------------
// One block = 16 rows. 8 waves; each wave owns a 16x64 quad of WMMA output
// tiles per iteration (A-fragment reused across 4 wmmas) and strides over the
// 128 column quads, maintaining a per-row top-10 in registers.
// Score monotone-equivalent to S = exp(-d2/gamma):  score = 2*g(i,j) - sq[j].
__global__ __launch_bounds__(256) void gram_topk_kernel(const float* __restrict__ X,
                                                        const float* __restrict__ sq,
                                                        int* __restrict__ knn) {
  __shared__ float sTile[8][16][64];          // 32 KB
  __shared__ float sMV[16][8][K_NEI];
  __shared__ int   sMI[16][8][K_NEI];

  const int lane = threadIdx.x & 31;
  const int wave = threadIdx.x >> 5;
  const int I    = blockIdx.x * 16;
  const int mrow = lane & 15;
  const int half = lane >> 4;

  float tv[K_NEI]; int ti[K_NEI];
#pragma unroll
  for (int q = 0; q < K_NEI; ++q) { tv[q] = -__builtin_inff(); ti[q] = -1; }

  const float* xa = X + (size_t)(I + mrow) * F_INN;   // A-frag row

  for (int qt = wave; qt < N_PTS / 64; qt += 8) {     // uniform: 16 iters/wave
    const int J = qt * 64;
    const float* xb0 = X + (size_t)(J + 0  + mrow) * F_INN;
    const float* xb1 = X + (size_t)(J + 16 + mrow) * F_INN;
    const float* xb2 = X + (size_t)(J + 32 + mrow) * F_INN;
    const float* xb3 = X + (size_t)(J + 48 + mrow) * F_INN;
    v8f c0 = {}, c1 = {}, c2 = {}, c3 = {};
#pragma unroll 2
    for (int k = 0; k < F_INN; k += 4) {
      v2f a  = *(const v2f*)(xa  + k + 2 * half);     // (k0,k1) low / (k2,k3) high
      v2f b0 = *(const v2f*)(xb0 + k + 2 * half);
      v2f b1 = *(const v2f*)(xb1 + k + 2 * half);
      v2f b2 = *(const v2f*)(xb2 + k + 2 * half);
      v2f b3 = *(const v2f*)(xb3 + k + 2 * half);
      c0 = __builtin_amdgcn_wmma_f32_16x16x4_f32(false, a, false, b0, (short)0, c0, false, false);
      c1 = __builtin_amdgcn_wmma_f32_16x16x4_f32(false, a, false, b1, (short)0, c1, false, false);
      c2 = __builtin_amdgcn_wmma_f32_16x16x4_f32(false, a, false, b2, (short)0, c2, false, false);
      c3 = __builtin_amdgcn_wmma_f32_16x16x4_f32(false, a, false, b3, (short)0, c3, false, false);
    }
    // D layout: vgpr v, lane L -> (row = v + 8*(L>>4), col = L&15)
#pragma unroll
    for (int v = 0; v < 8; ++v) {
      sTile[wave][v + 8 * half][mrow +  0] = c0[v];
      sTile[wave][v + 8 * half][mrow + 16] = c1[v];
      sTile[wave][v + 8 * half][mrow + 32] = c2[v];
      sTile[wave][v + 8 * half][mrow + 48] = c3[v];
    }
    __syncthreads();
    if (lane < 16) {
      const int i = I + lane;
#pragma unroll
      for (int cc = 0; cc < 64; ++cc) {
        int j = J + cc;
        float s = 2.0f * sTile[wave][lane][cc] - sq[j];
        if (j == i) s = -__builtin_inff();
        if (s > tv[K_NEI - 1]) {
          tv[K_NEI - 1] = s; ti[K_NEI - 1] = j;
#pragma unroll
          for (int q = K_NEI - 1; q > 0; --q) {
            if (tv[q] > tv[q - 1]) {
              float fv = tv[q]; tv[q] = tv[q - 1]; tv[q - 1] = fv;
              int iv = ti[q]; ti[q] = ti[q - 1]; ti[q - 1] = iv;
            }
          }
        }
      }
    }
    __syncthreads();
  }

  // merge 8 per-wave top-10 lists per row
  if (lane < 16) {
#pragma unroll
    for (int q = 0; q < K_NEI; ++q) { sMV[lane][wave][q] = tv[q]; sMI[lane][wave][q] = ti[q]; }
  }
  __syncthreads();
  if (threadIdx.x < 16) {
    int r = threadIdx.x;
    float bv[K_NEI]; int bi[K_NEI];
#pragma unroll
    for (int q = 0; q < K_NEI; ++q) { bv[q] = -__builtin_inff(); bi[q] = -1; }
    for (int w = 0; w < 8; ++w) {
#pragma unroll
      for (int q = 0; q < K_NEI; ++q) {
        float s = sMV[r][w][q]; int j = sMI[r][w][q];
        if (s > bv[K_NEI - 1]) {
          bv[K_NEI - 1] = s; bi[K_NEI - 1] = j;
#pragma unroll
          for (int p = K_NEI - 1; p > 0; --p) {
            if (bv[p] > bv[p - 1]) {
              float fv = bv[p]; bv[p] = bv[p - 1]; bv[p - 1] = fv;
              int iv = bi[p]; bi[p] = bi[p - 1]; bi[p - 1] = iv;
            }
          }
        }
      }
    }
#pragma unroll
    for (int q = 0; q < K_NEI; ++q) knn[(size_t)(I + r) * K_NEI + q] = bi[q];
  }
}

// -------------------- symmetrized adjacency bitset --------------------
__global__ __launch_bounds__(256) void scatter_kernel(const int* __restrict__ knn,
                                                      unsigned* __restrict__ mask) {
  int t = blockIdx.x * 256 + threadIdx.x;
  if (t >= N_PTS * K_NEI) return;
  int i = t / K_NEI;
  int j = knn[t];
  atomicOr(&mask[(size_t)i * MASKW + (j >> 5)], 1u << (j & 31));
  atomicOr(&mask[(size_t)j * MASKW + (i >> 5)], 1u << (i & 31));
}

// -------------------- degree / inverse degree / edge count --------------------
__global__ __launch_bounds__(256) void deg_kernel(const unsigned* __restrict__ mask,
                                                  float* __restrict__ deg,
                                                  float* __restrict__ invd,
                                                  float* __restrict__ accum) {
  int i = blockIdx.x * 256 + threadIdx.x;
  if (i >= N_PTS) return;
  int cnt = 0;
  for (int w = 0; w < MASKW; ++w) cnt += __popc(mask[(size_t)i * MASKW + w]);
  deg[i]  = (float)cnt;
  invd[i] = cnt > 0 ? 1.0f / (float)cnt : 0.0f;
  atomicAdd(&accum[1], (float)cnt);
}

// ---- fp32 WMMA GEMM: C[M,N] = A[M,K] @ B[K,N], 16x64 output per wave ----
__global__ __launch_bounds__(256) void wmma_gemm_kernel(const float* __restrict__ A,
                                                        const float* __restrict__ B,
                                                        float* __restrict__ C,
                                                        int M, int N, int K) {
  const int lane = threadIdx.x & 31;
  const int wave = threadIdx.x >> 5;
  const int quadsN = N >> 6;                       // N/64 quad-tiles per row band
  const int tile = blockIdx.x * 8 + wave;
  if (tile >= (M >> 4) * quadsN) return;           // wave-uniform
  const int tm = tile / quadsN, tq = tile % quadsN;
  const int mrow = lane & 15, half = lane >> 4;
  const float* aRow = A + (size_t)(tm * 16 + mrow) * K;
  const int col = tq * 64 + mrow;
  v8f c0 = {}, c1 = {}, c2 = {}, c3 = {};
#pragma unroll 2
  for (int k = 0; k < K; k += 4) {
    v2f a = *(const v2f*)(aRow + k + 2 * half);
    const float* br0 = B + (size_t)(k + 2 * half) * N + col;
    const float* br1 = br0 + N;
    v2f b0, b1, b2, b3;
    b0.x = br0[0];  b0.y = br1[0];
    b1.x = br0[16]; b1.y = br1[16];
    b2.x = br0[32]; b2.y = br1[32];
    b3.x = br0[48]; b3.y = br1[48];
    c0 = __builtin_amdgcn_wmma_f32_16x16x4_f32(false, a, false, b0, (short)0, c0, false, false);
    c1 = __builtin_amdgcn_wmma_f32_16x16x4_f32(false, a, false, b1, (short)0, c1, false, false);
    c2 = __builtin_amdgcn_wmma_f32_16x16x4_f32(false, a, false, b2, (short)0, c2, false, false);
    c3 = __builtin_amdgcn_wmma_f32_16x16x4_f32(false, a, false, b3, (short)0, c3, false, false);
  }
#pragma unroll
  for (int v = 0; v < 8; ++v) {
    float* crow = C + (size_t)(tm * 16 + v + 8 * half) * N + col;
    crow[0]  = c0[v];
    crow[16] = c1[v];
    crow[32] = c2[v];
    crow[48] = c3[v];
  }
}

// ---------- sparse aggregation + skip + bias + activation ----------
// out[i,:] = T[i,:]*skip + inv[i]*sum_{j in N(i)} T[j,:] + b ; act 0: relu(selu), 1: id
__global__ __launch_bounds__(256) void agg_kernel(const float* __restrict__ T,
                                                  const unsigned* __restrict__ mask,
                                                  const float* __restrict__ invd,
                                                  const float* __restrict__ skip,
                                                  const float* __restrict__ bias,
                                                  float* __restrict__ out,
                                                  int F, int act) {
  __shared__ unsigned smask[MASKW];
  const int i = blockIdx.x;
  smask[threadIdx.x] = mask[(size_t)i * MASKW + threadIdx.x];
  __syncthreads();
  const float inv = invd[i];
  for (int col = threadIdx.x; col < F; col += 256) {
    float acc = 0.f;
    for (int w = 0; w < MASKW; ++w) {
      unsigned bits = smask[w];
      while (bits) {
        int b = __ffs(bits) - 1;
        bits &= bits - 1;
        acc += T[(size_t)((w << 5) + b) * F + col];
      }
    }
    float v = T[(size_t)i * F + col] * skip[col] + inv * acc + bias[col];
    if (act == 0) {
      const float sc = 1.0507009873554805f, al = 1.6732632423543772f;
      float s = v > 0.f ? sc * v : sc * al * (expf(v) - 1.f);
      v = fmaxf(s, 0.f);   // relu(selu(v))
    }
    out[(size_t)i * F + col] = v;
  }
}

// -------------------- softmax(emb @ Wt^T + bt) --------------------
__global__ __launch_bounds__(256) void assign_kernel(const float* __restrict__ emb,
                                                     const float* __restrict__ Wt,
                                                     const float* __restrict__ bt,
                                                     float* __restrict__ assign) {
  int i = blockIdx.x * 256 + threadIdx.x;
  if (i >= N_PTS) return;
  const float* e = emb + (size_t)i * F_OUTT;
  float lg[N_CLUU];
  float mx = -__builtin_inff();
#pragma unroll
  for (int c = 0; c < N_CLUU; ++c) {
    const float* w = Wt + (size_t)c * F_OUTT;
    float acc = bt[c];
    for (int k = 0; k < F_OUTT; ++k) acc = fmaf(e[k], w[k], acc);
    lg[c] = acc;
    mx = fmaxf(mx, acc);
  }
  float sum = 0.f;
#pragma unroll
  for (int c = 0; c < N_CLUU; ++c) { lg[c] = expf(lg[c] - mx); sum += lg[c]; }
  float r = 1.0f / sum;
#pragma unroll
  for (int c = 0; c < N_CLUU; ++c) assign[(size_t)i * N_CLUU + c] = lg[c] * r;
}

// ---------- trace accumulation: one wave per row ----------
// traceGP += sum_{j in N(i)} <assign_i, assign_j>;  m[c] += assign[i,c]*deg_i
__global__ __launch_bounds__(32) void loss_kernel(const unsigned* __restrict__ mask,
                                                  const float* __restrict__ assign,
                                                  const float* __restrict__ deg,
                                                  float* __restrict__ accum) {
  const int i = blockIdx.x;
  const int lane = threadIdx.x;
  float ai[N_CLUU];
#pragma unroll
  for (int c = 0; c < N_CLUU; ++c) ai[c] = assign[(size_t)i * N_CLUU + c];
  float part = 0.f;
  for (int w = lane; w < MASKW; w += 32) {
    unsigned bits = mask[(size_t)i * MASKW + w];
    while (bits) {
      int b = __ffs(bits) - 1;
      bits &= bits - 1;
      const float* aj = assign + (size_t)((w << 5) + b) * N_CLUU;
      float d = 0.f;
#pragma unroll
      for (int c = 0; c < N_CLUU; ++c) d = fmaf(ai[c], aj[c], d);
      part += d;
    }
  }
#pragma unroll
  for (int off = 16; off > 0; off >>= 1) part += __shfl_down(part, off, 32);
  if (lane == 0) atomicAdd(&accum[0], part);
  if (lane < N_CLUU) atomicAdd(&accum[2 + lane], ai[lane] * deg[i]);
}

__global__ void finalize_kernel(const float* __restrict__ accum, float* __restrict__ out) {
  if (threadIdx.x == 0 && blockIdx.x == 0) {
    float E = accum[1];                 // n_edges = sum(deg)
    float tGP = accum[0];
    float tN = 0.f;
    for (int c = 0; c < N_CLUU; ++c) tN += accum[2 + c] * accum[2 + c];
    tN = tN / (2.0f * E);
    out[0] = -(tGP - tN) / (2.0f * E);
  }
}

// ==================== host launcher ====================
extern "C" void kernel_launch(void* const* d_in, const int* in_sizes, int n_in,
                              void* d_out, int out_size, void* d_ws, size_t ws_size,
                              hipStream_t stream) {
  const float* x     = (const float*)d_in[0];
  const float* W1    = (const float*)d_in[1];
  const float* b1    = (const float*)d_in[2];
  const float* skip1 = (const float*)d_in[3];
  const float* W2    = (const float*)d_in[4];
  const float* b2    = (const float*)d_in[5];
  const float* skip2 = (const float*)d_in[6];
  const float* Wt    = (const float*)d_in[7];
  const float* bt    = (const float*)d_in[8];
  float* out = (float*)d_out;

  float*    ws   = (float*)d_ws;
  float*    sq   = ws + OFF_SQ;
  float*    deg  = ws + OFF_DEG;
  float*    inv  = ws + OFF_INV;
  int*      knn  = (int*)(ws + OFF_KNN);
  unsigned* mask = (unsigned*)(ws + OFF_MASK);
  float*    acc  = ws + OFF_ACC;
  float*    T1   = ws + OFF_T1;
  float*    h    = ws + OFF_H;
  float*    T2   = ws + OFF_T2;
  float*    emb  = ws + OFF_EMB;
  float*    assign = ws + OFF_ASSIGN;

  // 1. zero bitmask + accumulators (contiguous)
  {
    size_t nz = (size_t)N_PTS * MASKW + 16;
    zero_kernel<<<(unsigned)((nz + 255) / 256), 256, 0, stream>>>(mask, nz);
  }
  // 2. row norms
  sq_kernel<<<N_PTS / 256, 256, 0, stream>>>(x, sq);
  // 3. fused gram + top-k (WMMA f32, 16x64 register blocking)
  gram_topk_kernel<<<N_PTS / 16, 256, 0, stream>>>(x, sq, knn);
  // 4. symmetrized adjacency bitset
  scatter_kernel<<<(N_PTS * K_NEI + 255) / 256, 256, 0, stream>>>(knn, mask);
  // 5. degrees, inverse degrees, n_edges
  deg_kernel<<<N_PTS / 256, 256, 0, stream>>>(mask, deg, inv, acc);
  // 6. T1 = x @ W1   (8192x512x512)
  wmma_gemm_kernel<<<(N_PTS / 16) * (F_HIDD / 64) / 8, 256, 0, stream>>>(
      x, W1, T1, N_PTS, F_HIDD, F_INN);
  // 7. h = relu(selu(T1*skip1 + A_hat@T1 + b1))
  agg_kernel<<<N_PTS, 256, 0, stream>>>(T1, mask, inv, skip1, b1, h, F_HIDD, 0);
  // 8. T2 = h @ W2   (8192x256x512)
  wmma_gemm_kernel<<<(N_PTS / 16) * (F_OUTT / 64) / 8, 256, 0, stream>>>(
      h, W2, T2, N_PTS, F_OUTT, F_HIDD);
  // 9. emb = T2*skip2 + A_hat@T2 + b2
  agg_kernel<<<N_PTS, 256, 0, stream>>>(T2, mask, inv, skip2, b2, emb, F_OUTT, 1);
  // 10. assign = softmax(emb @ Wt^T + bt)
  assign_kernel<<<N_PTS / 256, 256, 0, stream>>>(emb, Wt, bt, assign);
  // 11. trace terms
  loss_kernel<<<N_PTS, 32, 0, stream>>>(mask, assign, deg, acc);
  // 12. scalar loss
  finalize_kernel<<<1, 1, 0, stream>>>(acc, out);
}